// BilateralSliceApply_75110388072939
// MI455X (gfx1250) — compile-verified
//
#include <hip/hip_runtime.h>

// ext-vector types matching the probe-confirmed clang-23 signature of
// __builtin_amdgcn_tensor_load_to_lds:
//   (u32x4 g0, i32x8 g1, i32x4 g2, i32x4 g3, i32x8 g4, i32 cpol)
typedef unsigned int v4u __attribute__((ext_vector_type(4)));
typedef int          v8i __attribute__((ext_vector_type(8)));
typedef int          v4i __attribute__((ext_vector_type(4)));
typedef float        v2f __attribute__((ext_vector_type(2)));

namespace {
constexpr int kB   = 4;
constexpr int kC   = 12;
constexpr int kGD  = 8;
constexpr int kGH  = 16;
constexpr int kGW  = 16;
constexpr int kH   = 1024;
constexpr int kW   = 1024;
constexpr int TILE = 32;   // 32x32 pixel tile -> constant (fx, fy) grid cell
}

__device__ __forceinline__ int iclamp(int v, int lo, int hi) {
  return v < lo ? lo : (v > hi ? hi : v);
}

__device__ __forceinline__ v2f pk_fma(v2f a, v2f b, v2f c) {
  return __builtin_elementwise_fma(a, b, c);   // -> v_pk_fma_f32 on gfx1250
}

__global__ __launch_bounds__(256, 4)
void bilateral_slice_apply_kernel(const float* __restrict__ grid,
                                  const float* __restrict__ guide,
                                  const float* __restrict__ image,
                                  float* __restrict__ out)
{
  // TDM destination: [ch][z][dy][dx]  (dim0=x fastest, as TDM writes LDS)
  __shared__ __align__(16) float sgrid[kC * kGD * 2 * 2];

  const int b   = blockIdx.z;
  const int tx0 = blockIdx.x * TILE;
  const int ty0 = blockIdx.y * TILE;

  // Tile-constant grid cell: fx = floor((x+0.5)/64 - 0.5) is constant on
  // [64k+32, 64k+95]; every 32-aligned 32-wide tile sits inside one span.
  const int fx = (tx0 == 0) ? -1 : ((tx0 - TILE) >> 6);
  const int fy = (ty0 == 0) ? -1 : ((ty0 - TILE) >> 6);
  const int ax = iclamp(fx, 0, kGW - 2);   // left corner actually loaded
  const int ay = iclamp(fy, 0, kGH - 2);

  // ---- One TDM tensor load stages the whole 384-float tile slice to LDS ----
  // 4D tile: dim0=x(2, contiguous), dim1=y(2, stride 16), dim2=z(8, stride 256),
  // dim3=ch(12, stride 2048). Issued by wave 0 only; tracked with TENSORcnt.
  if (threadIdx.x < 32u) {
    const float* gbase =
        grid + ((size_t)b * (kC * kGD * kGH * kGW) + (size_t)ay * kGW + (size_t)ax);
    unsigned long long ga = (unsigned long long)gbase;
    unsigned lds_off = (unsigned)(unsigned long long)(void*)sgrid; // addr[31:0] = LDS offset

    v4u g0 = { 1u,                                   // count=1 (valid), user mode
               lds_off,                              // lds_addr (bytes)
               (unsigned)(ga & 0xFFFFFFFFull),       // global_addr[31:0]
               (unsigned)((ga >> 32) & 0x01FFFFFFull) | (2u << 30) }; // addr[56:32] | type=2

    v8i g1 = { (int)(2u << 16),                      // data_size=2 (4B); wg_mask=0
               (int)(((unsigned)kGW & 0xFFFFu) << 16),        // tensor_dim0 lo16 = 16
               (int)(((unsigned)kGH & 0xFFFFu) << 16),        // td0 hi | tensor_dim1 lo16 = 16
               (int)(2u << 16),                               // td1 hi | tile_dim0 = 2 (x)
               (int)(2u | ((unsigned)kGD << 16)),             // tile_dim1=2 (y) | tile_dim2=8 (z)
               kGW,                                           // tensor_dim0_stride = 16 (row)
               (int)(((unsigned)(kGH * kGW) & 0xFFFFu) << 16),// s0 hi | tensor_dim1_stride lo16 = 256
               0 };                                           // tensor_dim1_stride hi

    v4i g2 = { kGD,                                  // tensor_dim2 = 8
               kC,                                   // tensor_dim3 = 12
               kGD * kGH * kGW,                      // tensor_dim2_stride = 2048 (ch)
               (int)((unsigned)kC << 16) };          // tile_dim3 = 12

    v4i g3 = { 0, 0, 0, 0 };                         // dim4 unused (tile_dim4=0)
    v8i g4 = { 0, 0, 0, 0, 0, 0, 0, 0 };             // extra VADDR group: NULL/zero

    __builtin_amdgcn_tensor_load_to_lds(g0, g1, g2, g3, g4, 0);
    __builtin_amdgcn_s_wait_tensorcnt(0);
  }

  // ---- Overlap: pull guide + image while the TDM runs ----
  const int t = (int)threadIdx.x;
  const int x = tx0 + (t & 7) * 4;      // 4 consecutive pixels per thread
  const int y = ty0 + (t >> 3);

  const size_t hw = (size_t)kH * kW;
  const float* gu = guide + (size_t)b * hw + (size_t)y * kW + x;
  const float* im = image + (size_t)b * 3 * hw + (size_t)y * kW + x;

  const float4 g4v = *(const float4*)gu;
  const float4 r4  = *(const float4*)(im);
  const float4 q4  = *(const float4*)(im + hw);
  const float4 b4  = *(const float4*)(im + 2 * hw);

  __syncthreads();   // LDS tile visible to all 8 waves

  const float4* __restrict__ lds4 = (const float4*)sgrid; // [ch*8 + z] -> 4 xy-corners

  // y hat weights (thread-constant), folded at tensor edges so they address
  // the loaded corners [ay, ay+1] exactly like the reference's clamp.
  const float gy  = ((float)y + 0.5f) * ((float)kGH / (float)kH);
  const float tyf = gy - 0.5f - (float)fy;
  const float wy0 = (fy < 0) ? 1.0f : ((fy > kGH - 2) ? 0.0f : (1.0f - tyf));
  const float wy1 = 1.0f - wy0;

  const float gzv[4] = { g4v.x, g4v.y, g4v.z, g4v.w };
  const float rv[4]  = { r4.x, r4.y, r4.z, r4.w };
  const float qv[4]  = { q4.x, q4.y, q4.z, q4.w };
  const float bv[4]  = { b4.x, b4.y, b4.z, b4.w };

  float o0[4], o1[4], o2[4];

#pragma unroll
  for (int p = 0; p < 4; ++p) {
    const float gx  = ((float)(x + p) + 0.5f) * ((float)kGW / (float)kW);
    const float txf = gx - 0.5f - (float)fx;
    const float wx0 = (fx < 0) ? 1.0f : ((fx > kGW - 2) ? 0.0f : (1.0f - txf));
    const float wx1 = 1.0f - wx0;

    const float gz  = gzv[p] * (float)kGD;
    const float fzf = __builtin_floorf(gz - 0.5f);
    const float tzf = gz - 0.5f - fzf;           // in [0,1)
    const int   iz  = (int)fzf;                  // in [-1, 7]
    const int   iz0 = iz < 0 ? 0 : iz;
    const int   iz1 = (iz + 1) > (kGD - 1) ? (kGD - 1) : (iz + 1);
    const float wz0 = 1.0f - tzf;
    const float wz1 = tzf;

    // Packed-f32 weight pairs: (dx0, dx1) per (dy, dz) combination.
    const v2f wxp = { wx0, wx1 };
    const v2f w01 = wxp * wy0;        // (w00, w01)
    const v2f w23 = wxp * wy1;        // (w10, w11)
    const v2f u01 = w01 * wz0, u23 = w23 * wz0;
    const v2f v01 = w01 * wz1, v23 = w23 * wz1;

    const v2f rr = { rv[p], rv[p] };
    const v2f qq = { qv[p], qv[p] };
    const v2f bb = { bv[p], bv[p] };

    float acc[3];
#pragma unroll
    for (int i = 0; i < 3; ++i) {
      v2f cp[4];
#pragma unroll
      for (int j = 0; j < 4; ++j) {
        const int ch = i * 4 + j;
        const float4 A  = lds4[ch * kGD + iz0];  // 4 xy-corners @ z = iz0
        const float4 Bq = lds4[ch * kGD + iz1];  // 4 xy-corners @ z = iz1
        const v2f a01 = { A.x,  A.y  }, a23 = { A.z,  A.w  };
        const v2f b01 = { Bq.x, Bq.y }, b23 = { Bq.z, Bq.w };
        v2f s = a01 * u01;                 // v_pk_mul_f32
        s = pk_fma(a23, u23, s);           // v_pk_fma_f32
        s = pk_fma(b01, v01, s);
        s = pk_fma(b23, v23, s);
        cp[j] = s;                         // pair accumulator (no horizontal add)
      }
      // Affine apply done pairwise; single horizontal add at the end.
      v2f t2 = cp[3];
      t2 = pk_fma(cp[2], bb, t2);
      t2 = pk_fma(cp[1], qq, t2);
      t2 = pk_fma(cp[0], rr, t2);
      acc[i] = t2.x + t2.y;
    }
    o0[p] = acc[0]; o1[p] = acc[1]; o2[p] = acc[2];
  }

  float* op = out + (size_t)b * 3 * hw + (size_t)y * kW + x;
  *(float4*)(op)          = make_float4(o0[0], o0[1], o0[2], o0[3]);
  *(float4*)(op + hw)     = make_float4(o1[0], o1[1], o1[2], o1[3]);
  *(float4*)(op + 2 * hw) = make_float4(o2[0], o2[1], o2[2], o2[3]);
}

extern "C" void kernel_launch(void* const* d_in, const int* in_sizes, int n_in,
                              void* d_out, int out_size, void* d_ws, size_t ws_size,
                              hipStream_t stream) {
  (void)in_sizes; (void)n_in; (void)out_size; (void)d_ws; (void)ws_size;
  const float* grid  = (const float*)d_in[0];
  const float* guide = (const float*)d_in[1];
  const float* image = (const float*)d_in[2];
  float* out = (float*)d_out;

  dim3 g(kW / TILE, kH / TILE, kB);   // (32, 32, 4)
  dim3 blk(256);
  hipLaunchKernelGGL(bilateral_slice_apply_kernel, g, blk, 0, stream,
                     grid, guide, image, out);
}